// GCNLayer_55439437857136
// MI455X (gfx1250) — compile-verified
//
#include <hip/hip_runtime.h>

typedef __attribute__((ext_vector_type(2))) float v2f;
typedef __attribute__((ext_vector_type(8))) float v8f;

#define N_NODES 100000
#define N_EDGES 1600000
#define DIM 128
#define EPSV 1e-9f

// ---------------------------------------------------------------- zero fill
__global__ __launch_bounds__(256) void zero_kernel(float4* __restrict__ p, long long n4) {
    long long i = (long long)blockIdx.x * 256 + threadIdx.x;
    if (i < n4) p[i] = make_float4(0.f, 0.f, 0.f, 0.f);
}

// ------------------------------------------------- edge scatter (u_mul_e + segment_sum)
// One wave per edge; lane l handles floats [4l, 4l+4). Edge index is forced
// into an SGPR so src/dst/w (and the optional D_norm[src] fold) become scalar
// constant-cache loads instead of 32x-replicated vector loads, and the
// gather/atomic addressing becomes SADDR+voffset. Both the gathered rows and
// the accumulator live in the 192MB L2, so this kernel is L2-atomic bound.
__global__ __launch_bounds__(256) void scatter_kernel(const float* __restrict__ x,
                                                      const int* __restrict__ src,
                                                      const int* __restrict__ dst,
                                                      const float* __restrict__ w,
                                                      const float* __restrict__ dn_src, // null hop1; D_norm hop2
                                                      float* __restrict__ agg) {
    long long g = (long long)blockIdx.x * 256 + threadIdx.x;
    int e    = __builtin_amdgcn_readfirstlane((int)(g >> 5));   // wave-uniform edge id
    int lane = threadIdx.x & 31;
    if (e >= N_EDGES) return;
    int   s  = src[e];                 // scalar loads (SGPR index)
    int   d  = dst[e];
    float we = w[e];
    if (dn_src) we *= dn_src[s];       // fold previous hop's D_norm into edge weight
    float4 v = ((const float4*)(x + (size_t)s * DIM))[lane];
    float* dp = agg + (size_t)d * DIM + lane * 4;
    atomicAdd(dp + 0, v.x * we);
    atomicAdd(dp + 1, v.y * we);
    atomicAdd(dp + 2, v.z * we);
    atomicAdd(dp + 3, v.w * we);
}

// ------------------------------- fused  (X*D_norm) @ W^T + b -> relu -> layernorm
// 8 waves / block, each wave computes a 16x128 output strip with
// v_wmma_f32_16x16x4_f32.  W[i] (64KB) staged in LDS.  D_norm (when non-null)
// is applied to the A fragment at load time, eliminating a separate pass.
__global__ __launch_bounds__(256) void gemm_ln_kernel(const float* __restrict__ X,
                                                      const float* __restrict__ dn,    // null for hop 0
                                                      const float* __restrict__ Wm,    // [128,128] row-major W[o][k]
                                                      const float* __restrict__ bias,  // [128]
                                                      const float* __restrict__ scl,   // [128]
                                                      const float* __restrict__ off,   // [128]
                                                      float* __restrict__ out,         // [N, 384]
                                                      int col_base) {
    __shared__ float sW[DIM * DIM];      // 64 KB (WGP has 320 KB)
    for (int t = threadIdx.x; t < DIM * DIM / 4; t += 256)
        ((float4*)sW)[t] = ((const float4*)Wm)[t];
    __syncthreads();

    const int wave = threadIdx.x >> 5;
    const int lane = threadIdx.x & 31;
    const int hi   = lane >> 4;          // half-wave selects K offset for A/B frags
    const int lm   = lane & 15;
    const int row0 = blockIdx.x * 128 + wave * 16;
    if (row0 >= N_NODES) return;         // wave-uniform exit, EXEC stays all-1 for WMMA

    // A fragment source: lane holds X[row0+lm][k + 2*hi + {0,1}]  (ISA 16x4 f32 layout)
    int arow = row0 + lm;
    if (arow >= N_NODES) arow = N_NODES - 1;   // clamp loads in the ragged last block
    const float* aptr = X + (size_t)arow * DIM + hi * 2;
    const float dsc = dn ? dn[arow] : 1.0f;    // fold h = agg * D_norm into A load

    v8f acc[8];
#pragma unroll
    for (int ct = 0; ct < 8; ++ct)
#pragma unroll
        for (int j = 0; j < 8; ++j) acc[ct][j] = 0.f;

    for (int k = 0; k < DIM; k += 4) {
        v2f a = *(const v2f*)(aptr + k);
        a.x *= dsc;
        a.y *= dsc;
#pragma unroll
        for (int ct = 0; ct < 8; ++ct) {
            // B fragment: B[k][n] = W[n][k]; lane holds W[ct*16+lm][k + 2*hi + {0,1}]
            v2f bf = *(const v2f*)(&sW[(ct * 16 + lm) * DIM + k + hi * 2]);
            acc[ct] = __builtin_amdgcn_wmma_f32_16x16x4_f32(
                false, a, false, bf, (short)0, acc[ct], false, false);
        }
    }

    // Epilogue: bias + relu, then per-row (m = j + 8*hi) mean/var across 128 cols.
    float s1[8], s2[8];
#pragma unroll
    for (int j = 0; j < 8; ++j) { s1[j] = 0.f; s2[j] = 0.f; }

#pragma unroll
    for (int ct = 0; ct < 8; ++ct) {
        float bn = bias[ct * 16 + lm];
#pragma unroll
        for (int j = 0; j < 8; ++j) {
            float x = acc[ct][j] + bn;
            x = fmaxf(x, 0.f);
            acc[ct][j] = x;
            s1[j] += x;
            s2[j] += x * x;
        }
    }
    // reduce across the 16 lanes holding this row (xor masks < 16 stay in-half on wave32)
#pragma unroll
    for (int j = 0; j < 8; ++j) {
#pragma unroll
        for (int m = 1; m < 16; m <<= 1) {
            s1[j] += __shfl_xor(s1[j], m, 32);
            s2[j] += __shfl_xor(s2[j], m, 32);
        }
    }
    float mean[8], rinv[8];
#pragma unroll
    for (int j = 0; j < 8; ++j) {
        float mu  = s1[j] * (1.0f / DIM);
        float var = s2[j] * (1.0f / DIM) - mu * mu + EPSV;   // ddof=0 + eps
        mean[j] = mu;
        rinv[j] = rsqrtf(var);
    }

#pragma unroll
    for (int ct = 0; ct < 8; ++ct) {
        int n = ct * 16 + lm;
        float sc = scl[n];
        float of = off[n];
#pragma unroll
        for (int j = 0; j < 8; ++j) {
            int row = row0 + j + 8 * hi;
            if (row < N_NODES) {
                float v = (acc[ct][j] - mean[j]) * sc * rinv[j] + of;
                // streamed once, never re-read: keep it out of the L2 working set
                __builtin_nontemporal_store(v, &out[(size_t)row * (3 * DIM) + col_base + n]);
            }
        }
    }
}

// ---------------------------------------------------------------------------
extern "C" void kernel_launch(void* const* d_in, const int* in_sizes, int n_in,
                              void* d_out, int out_size, void* d_ws, size_t ws_size,
                              hipStream_t stream) {
    const float* features = (const float*)d_in[0];
    const int*   src      = (const int*)d_in[1];
    const int*   dst      = (const int*)d_in[2];
    const float* w        = (const float*)d_in[3];
    const float* D_norm   = (const float*)d_in[4];
    const float* W        = (const float*)d_in[5];   // [3,128,128]
    const float* b        = (const float*)d_in[6];   // [3,128]
    const float* scale    = (const float*)d_in[7];   // [3,128]
    const float* offset   = (const float*)d_in[8];   // [3,128]
    float*       out      = (float*)d_out;           // [N, 384]

    float* a1 = (float*)d_ws;                        // [N,128] hop-1 raw aggregate
    float* a2 = a1 + (size_t)N_NODES * DIM;          // [N,128] hop-2 raw aggregate

    const long long n4 = (long long)N_NODES * DIM / 4;
    const int zb = (int)((n4 + 255) / 256);
    const long long st = (long long)N_EDGES * 32;
    const int sb = (int)((st + 255) / 256);
    const int gb = (N_NODES + 127) / 128;

    // hop 1: a1 = segsum(features[src] * w)          (h1 = a1 * D_norm, applied lazily)
    zero_kernel   <<<zb, 256, 0, stream>>>((float4*)a1, n4);
    scatter_kernel<<<sb, 256, 0, stream>>>(features, src, dst, w, nullptr, a1);
    // hop 2: a2 = segsum(a1[src] * (D_norm[src]*w))  (h2 = a2 * D_norm, applied lazily)
    zero_kernel   <<<zb, 256, 0, stream>>>((float4*)a2, n4);
    scatter_kernel<<<sb, 256, 0, stream>>>(a1, src, dst, w, D_norm, a2);
    // fused transform + layernorm for each hop, writing concat slots
    gemm_ln_kernel<<<gb, 256, 0, stream>>>(features, nullptr, W,             b,         scale,         offset,         out, 0);
    gemm_ln_kernel<<<gb, 256, 0, stream>>>(a1, D_norm, W + DIM * DIM,   b + DIM,   scale + DIM,   offset + DIM,   out, DIM);
    gemm_ln_kernel<<<gb, 256, 0, stream>>>(a2, D_norm, W + 2 * DIM*DIM, b + 2*DIM, scale + 2*DIM, offset + 2*DIM, out, 2 * DIM);
}